// NBVHModel_25821343383809
// MI455X (gfx1250) — compile-verified
//
#include <hip/hip_runtime.h>
#include <hip/hip_bf16.h>

typedef __attribute__((ext_vector_type(2))) float v2f;
typedef __attribute__((ext_vector_type(8))) float v8f;

#define TABLE_MASK 0x7FFFFu   // TABLE_SIZE = 2^19
#define ENC_DEPTH 8
#define N_POINTS 16
#define ENC_DIM 16

// (c+1) * PIS[c], low 32 bits — only low 19 bits of the xor survive the mod.
__device__ constexpr unsigned kc(int c) {
    constexpr unsigned long long P[8] = {
        774363409ull, 2654435761ull, 805459861ull, 100000007ull,
        334363391ull, 1334363413ull, 734363407ull, 2134363393ull};
    return (unsigned)(((unsigned long long)(c + 1)) * P[c]);
}

// Reference corner ordering:
// w0=(1-x)(1-y)(1-z) w1=x(1-y)(1-z) w2=(1-x)y(1-z) w3=(1-x)(1-y)z
// w4=x(1-y)z         w5=(1-x)yz     w6=xy(1-z)     w7=xyz
__device__ __forceinline__ float corner_w(int c, float x, float y, float z) {
    const unsigned UX = 0xD2u; // bits {1,4,6,7}
    const unsigned UY = 0xE4u; // bits {2,5,6,7}
    const unsigned UZ = 0xB8u; // bits {3,4,5,7}
    float fx = ((UX >> c) & 1u) ? x : 1.0f - x;
    float fy = ((UY >> c) & 1u) ? y : 1.0f - y;
    float fz = ((UZ >> c) & 1u) ? z : 1.0f - z;
    return fx * fy * fz;
}

__global__ void ws_init_kernel(int* ws) {
    if (blockIdx.x == 0 && threadIdx.x == 0) ws[0] = 0;
}

__global__ void depth_max_kernel(const int* __restrict__ depth, int n, int* __restrict__ ws) {
    __shared__ int smax;
    if (threadIdx.x == 0) smax = 0;
    __syncthreads();
    int i = blockIdx.x * blockDim.x + threadIdx.x;
    int v = 0;
    for (; i < n; i += gridDim.x * blockDim.x) v = max(v, depth[i]);
    atomicMax(&smax, v);
    __syncthreads();
    if (threadIdx.x == 0) atomicMax(ws, smax);
}

// One wave32 per (ray, level) tile. Block = 256 threads = 8 waves = 8 levels of one ray.
__global__ __launch_bounds__(256) void nbvh_enc_kernel(
    const float* __restrict__ inp,          // [R,16,3]
    const int*   __restrict__ history,      // [R,64]
    const float* __restrict__ nodes_min,    // [N,3]
    const float* __restrict__ nodes_extent, // [N,3]
    const float* __restrict__ emb,          // [2^19,16]
    const int*   __restrict__ ws,           // ws[0] = max(depth)
    float*       __restrict__ out)          // [R, 8*16*16]
{
    const int r    = blockIdx.x;
    const int l    = threadIdx.x >> 5;   // wave id = level 0..7
    const int lane = threadIdx.x & 31;
    const int half = lane >> 4;          // 0: K=0,1 / rows 0..7 ; 1: K=2,3 / rows 8..15
    const int m    = lane & 15;          // point id (A rows) == dim id (B cols)

    const int idx = history[r * 64 + l];
    const unsigned uidx = (unsigned)idx;

    // Node AABB (uniform across wave; broadcast load from L2)
    const float pmx = nodes_min[idx * 3 + 0];
    const float pmy = nodes_min[idx * 3 + 1];
    const float pmz = nodes_min[idx * 3 + 2];
    const float exx = nodes_extent[idx * 3 + 0];
    const float exy = nodes_extent[idx * 3 + 1];
    const float exz = nodes_extent[idx * 3 + 2];

    // Point m of ray r, normalized + clamped
    const int pbase = (r * N_POINTS + m) * 3;
    float x = fminf(fmaxf((inp[pbase + 0] - pmx) / exx, 0.0f), 1.0f);
    float y = fminf(fmaxf((inp[pbase + 1] - pmy) / exy, 0.0f), 1.0f);
    float z = fminf(fmaxf((inp[pbase + 2] - pmz) / exz, 0.0f), 1.0f);

    // Corner assignment per WMMA f32 16x16x4 layout:
    // first WMMA: lanes 0-15 -> corners 0,1 ; lanes 16-31 -> corners 2,3
    // second WMMA: +4
    const int c0 = 2 * half;
    const int c1 = c0 + 1;
    const int c2 = c0 + 4;
    const int c3 = c0 + 5;

    // A-matrix: trilinear weights
    v2f a01, a23;
    a01.x = corner_w(c0, x, y, z);
    a01.y = corner_w(c1, x, y, z);
    a23.x = corner_w(c2, x, y, z);
    a23.y = corner_w(c3, x, y, z);

    // B-matrix: gathered embedding rows (each half-wave reads one 64B row)
    const unsigned k0 = half ? kc(2) : kc(0);
    const unsigned k1 = half ? kc(3) : kc(1);
    const unsigned k2 = half ? kc(6) : kc(4);
    const unsigned k3 = half ? kc(7) : kc(5);
    const unsigned t0 = (uidx ^ k0) & TABLE_MASK;
    const unsigned t1 = (uidx ^ k1) & TABLE_MASK;
    const unsigned t2 = (uidx ^ k2) & TABLE_MASK;
    const unsigned t3 = (uidx ^ k3) & TABLE_MASK;
    v2f b01, b23;
    b01.x = emb[t0 * ENC_DIM + m];
    b01.y = emb[t1 * ENC_DIM + m];
    b23.x = emb[t2 * ENC_DIM + m];
    b23.y = emb[t3 * ENC_DIM + m];

    // D = A(16x8) * B(8x16) via two f32 16x16x4 WMMAs (exact f32, K=4 each)
    v8f acc = {};
    acc = __builtin_amdgcn_wmma_f32_16x16x4_f32(false, a01, false, b01,
                                                (short)0, acc, false, false);
    acc = __builtin_amdgcn_wmma_f32_16x16x4_f32(false, a23, false, b23,
                                                (short)0, acc, false, false);

    // active[l] = l < min(max(depth), 8); write zeros for inactive (out is poisoned)
    const int   lmax  = min(ws[0], ENC_DEPTH);
    const float scale = (l < lmax) ? 1.0f : 0.0f;

    // D layout: VGPR j = row (j + 8*half), col = m. 8 stores, 64B-coalesced.
    float* outp = out + ((size_t)(r * ENC_DEPTH + l)) * (N_POINTS * ENC_DIM);
#pragma unroll
    for (int j = 0; j < 8; ++j) {
        const int row = j + half * 8;
        outp[row * ENC_DIM + m] = acc[j] * scale;
    }
}

extern "C" void kernel_launch(void* const* d_in, const int* in_sizes, int n_in,
                              void* d_out, int out_size, void* d_ws, size_t ws_size,
                              hipStream_t stream) {
    const float* inp          = (const float*)d_in[0];
    const int*   history      = (const int*)d_in[1];
    const int*   depth        = (const int*)d_in[2];
    const float* nodes_min    = (const float*)d_in[3];
    const float* nodes_extent = (const float*)d_in[4];
    const float* emb          = (const float*)d_in[5];
    float* out = (float*)d_out;
    int*   ws  = (int*)d_ws;

    const int n_rays = in_sizes[2]; // depth has one entry per ray

    ws_init_kernel<<<1, 64, 0, stream>>>(ws);
    depth_max_kernel<<<128, 256, 0, stream>>>(depth, n_rays, ws);
    nbvh_enc_kernel<<<n_rays, 256, 0, stream>>>(inp, history, nodes_min,
                                                nodes_extent, emb, ws, out);
}